// VanillaGNNLayer_38070590112568
// MI455X (gfx1250) — compile-verified
//
#include <hip/hip_runtime.h>
#include <hip/hip_bf16.h>

// Problem constants (fixed by the reference).
#define N_NODES 8192
#define N_EDGES 262144
#define DIM_IN  512
#define DIM_OUT 256

typedef float v2f __attribute__((ext_vector_type(2)));
typedef float v8f __attribute__((ext_vector_type(8)));

// Device-scope fp32 atomic add that is guaranteed to lower to the CDNA5
// hardware instruction (no CAS fallback regardless of fp-atomic compile mode).
__device__ __forceinline__ void atomic_add_f32_dev(float* p, float v) {
    asm volatile("global_atomic_add_f32 %0, %1, off scope:SCOPE_DEV"
                 :
                 : "v"(p), "v"(v)
                 : "memory");
}

// ---------------------------------------------------------------------------
// Phase 1: degree. deg[i] = 1 + sum_{e: src==i} w[e]
// ---------------------------------------------------------------------------
__global__ void init_deg_kernel(float* __restrict__ deg) {
    int i = blockIdx.x * blockDim.x + threadIdx.x;
    if (i < N_NODES) deg[i] = 1.0f;
}

__global__ void deg_scatter_kernel(const int* __restrict__ edge_index,
                                   const float* __restrict__ edge_weight,
                                   float* __restrict__ deg) {
    int e = blockIdx.x * blockDim.x + threadIdx.x;
    if (e < N_EDGES) {
        int s = edge_index[e];  // row 0 of [2, E]
        atomic_add_f32_dev(deg + s, edge_weight[e]);
    }
}

__global__ void rsqrt_kernel(float* __restrict__ deg) {
    int i = blockIdx.x * blockDim.x + threadIdx.x;
    if (i < N_NODES) deg[i] = 1.0f / sqrtf(deg[i]);
}

// ---------------------------------------------------------------------------
// Phase 2: h_s[m, n] = dinv[m] * sum_k x[m,k] * W[n,k]
// One wave32 computes a 16(M) x 64(N) tile with V_WMMA_F32_16X16X4_F32 and
// 4 accumulators, so each K-step is 1 A-load + 4 B-loads + 4 WMMAs
// (A fragment reused across the 4 N-subtiles -> 1.25 loads per WMMA).
//
// Fragment layouts (ISA 7.12.2, f32):
//   A (16x4):  lanes 0-15 hold M=0..15; VGPR0/1 = K {0,1}; lanes 16-31 = K {2,3}
//   B (4x16):  same pattern with N across lanes
//   C/D (16x16, 8 VGPRs): N = lane&15, M = 8*(lane>>4) + vgpr_index
// ---------------------------------------------------------------------------
__global__ void gemm_wmma_scale_kernel(const float* __restrict__ x,
                                       const float* __restrict__ W,
                                       const float* __restrict__ dinv,
                                       float* __restrict__ hs) {
    const int wave = threadIdx.x >> 5;            // 0..7
    const int lane = threadIdx.x & 31;
    const int half = lane >> 4;                   // 0 or 1
    const int lrow = lane & 15;

    const int m0 = (blockIdx.x * 2 + (wave >> 2)) * 16;  // 512 m-tiles
    const int n0 = (wave & 3) * 64;                      // 4 n-groups of 64

    const float2* __restrict__ xa =
        (const float2*)(x + (size_t)(m0 + lrow) * DIM_IN);
    const float2* __restrict__ wb0 =
        (const float2*)(W + (size_t)(n0 +  0 + lrow) * DIM_IN);
    const float2* __restrict__ wb1 =
        (const float2*)(W + (size_t)(n0 + 16 + lrow) * DIM_IN);
    const float2* __restrict__ wb2 =
        (const float2*)(W + (size_t)(n0 + 32 + lrow) * DIM_IN);
    const float2* __restrict__ wb3 =
        (const float2*)(W + (size_t)(n0 + 48 + lrow) * DIM_IN);

    v8f acc0 = {}, acc1 = {}, acc2 = {}, acc3 = {};
#pragma unroll 4
    for (int kk = 0; kk < DIM_IN / 4; ++kk) {
        const int fi = kk * 2 + half;             // K = 4*kk + 2*half + {0,1}
        float2 av = xa[fi];
        float2 b0 = wb0[fi];
        float2 b1 = wb1[fi];
        float2 b2 = wb2[fi];
        float2 b3 = wb3[fi];
        v2f a;  a[0]  = av.x; a[1]  = av.y;
        v2f v0; v0[0] = b0.x; v0[1] = b0.y;
        v2f v1; v1[0] = b1.x; v1[1] = b1.y;
        v2f v2; v2[0] = b2.x; v2[1] = b2.y;
        v2f v3; v3[0] = b3.x; v3[1] = b3.y;
        acc0 = __builtin_amdgcn_wmma_f32_16x16x4_f32(false, a, false, v0,
                                                     (short)0, acc0, false, false);
        acc1 = __builtin_amdgcn_wmma_f32_16x16x4_f32(false, a, false, v1,
                                                     (short)0, acc1, false, false);
        acc2 = __builtin_amdgcn_wmma_f32_16x16x4_f32(false, a, false, v2,
                                                     (short)0, acc2, false, false);
        acc3 = __builtin_amdgcn_wmma_f32_16x16x4_f32(false, a, false, v3,
                                                     (short)0, acc3, false, false);
    }

#pragma unroll
    for (int r = 0; r < 8; ++r) {
        const int m = m0 + 8 * half + r;
        const float dv = dinv[m];
        float* __restrict__ row = hs + (size_t)m * DIM_OUT + lrow;
        row[n0 +  0] = acc0[r] * dv;
        row[n0 + 16] = acc1[r] * dv;
        row[n0 + 32] = acc2[r] * dv;
        row[n0 + 48] = acc3[r] * dv;
    }
}

// ---------------------------------------------------------------------------
// Phase 3: out[i,:] = dinv[i] * h_s[i,:]   (identity/self-loop term, with the
// output-side D^{-1/2} already folded in)
// ---------------------------------------------------------------------------
__global__ void init_out_kernel(const float* __restrict__ hs,
                                const float* __restrict__ dinv,
                                float* __restrict__ out) {
    int row = blockIdx.x;                 // 8192 rows
    int t   = threadIdx.x;                // 256 columns
    size_t idx = (size_t)row * DIM_OUT + t;
    out[idx] = dinv[row] * hs[idx];
}

// ---------------------------------------------------------------------------
// Phase 4: per-edge scatter: out[dst,:] += (w * dinv[dst]) * h_s[src,:]
// One block (256 threads) per edge; column t per thread -> fully coalesced
// loads of h_s[src] and coalesced fp32 atomics into out[dst] (resolve in L2).
// ---------------------------------------------------------------------------
__global__ void edge_scatter_kernel(const int* __restrict__ edge_index,
                                    const float* __restrict__ edge_weight,
                                    const float* __restrict__ dinv,
                                    const float* __restrict__ hs,
                                    float* __restrict__ out) {
    int e = blockIdx.x;
    int t = threadIdx.x;
    int s = edge_index[e];
    int d = edge_index[N_EDGES + e];
    float coef = edge_weight[e] * dinv[d];
    float v = coef * hs[(size_t)s * DIM_OUT + t];
    atomic_add_f32_dev(out + (size_t)d * DIM_OUT + t, v);
}

// ---------------------------------------------------------------------------
extern "C" void kernel_launch(void* const* d_in, const int* in_sizes, int n_in,
                              void* d_out, int out_size, void* d_ws, size_t ws_size,
                              hipStream_t stream) {
    (void)in_sizes; (void)n_in; (void)out_size; (void)ws_size;

    const float* x  = (const float*)d_in[0];       // [N, DIN]
    const int*   ei = (const int*)d_in[1];         // [2, E]
    const float* ew = (const float*)d_in[2];       // [E]
    const float* W  = (const float*)d_in[3];       // [DOUT, DIN]
    float* out = (float*)d_out;                    // [N, DOUT]

    // Workspace layout: deg/dinv (N floats, in place), then h_s (N*DOUT floats)
    float* deg = (float*)d_ws;
    float* hs  = (float*)((char*)d_ws + 32768);    // 8192*4 = 32768 bytes

    // Phase 1: degree + rsqrt
    init_deg_kernel<<<dim3(N_NODES / 256), dim3(256), 0, stream>>>(deg);
    deg_scatter_kernel<<<dim3(N_EDGES / 256), dim3(256), 0, stream>>>(ei, ew, deg);
    rsqrt_kernel<<<dim3(N_NODES / 256), dim3(256), 0, stream>>>(deg);

    // Phase 2: h_s = rowscale(dinv) * (x @ W^T) via f32 WMMA.
    // 2048 waves total: 256 blocks x 8 waves; each wave owns a 16x64 tile.
    gemm_wmma_scale_kernel<<<dim3(N_NODES / 32), dim3(256), 0, stream>>>(
        x, W, deg, hs);

    // Phase 3: self-loop init (output-side dinv folded in)
    init_out_kernel<<<dim3(N_NODES), dim3(256), 0, stream>>>(hs, deg, out);

    // Phase 4: edge aggregation with device-scope fp32 atomics
    edge_scatter_kernel<<<dim3(N_EDGES), dim3(256), 0, stream>>>(
        ei, ew, deg, hs, out);
}